// TaskAlignedAssigner_89850715832993
// MI455X (gfx1250) — compile-verified
//
#include <hip/hip_runtime.h>
#include <hip/hip_bf16.h>

typedef __attribute__((ext_vector_type(2))) float v2f;
typedef __attribute__((ext_vector_type(8))) float v8f;

#define TAA_NUM_CLASSES 80
#define MAXM 32

__device__ __forceinline__ float ov_iou(float px1, float py1, float px2, float py2,
                                        float gx1, float gy1, float gx2, float gy2) {
  float w1 = px2 - px1, h1 = (py2 - py1) + 1e-7f;
  float w2 = gx2 - gx1, h2 = (gy2 - gy1) + 1e-7f;
  float iw = fmaxf(fminf(px2, gx2) - fmaxf(px1, gx1), 0.f);
  float ih = fmaxf(fminf(py2, gy2) - fmaxf(py1, gy1), 0.f);
  float inter = iw * ih;
  float uni = w1 * h1 + w2 * h2 - inter + 1e-7f;
  return fmaxf(inter / uni, 0.f);
}

// argmax-over-anchors via u64 atomicMax; tie-break to LOWEST anchor like jnp.argmax
__device__ __forceinline__ unsigned long long packmax(float v, int a) {
  return ((unsigned long long)__float_as_uint(v) << 32) |
         (unsigned)(0x7FFFFFFF - a);
}
__device__ __forceinline__ int unpack_anchor(unsigned long long p) {
  return 0x7FFFFFFF - (int)(unsigned)(p & 0xFFFFFFFFull);
}

// ---------------- K0: zero small reduction buffers ----------------
__global__ void k0_init(unsigned long long* g_am, unsigned long long* g_ap,
                        unsigned* g_any, unsigned* g_hasin, unsigned* g_cnt, int n) {
  int i = blockIdx.x * blockDim.x + threadIdx.x;
  if (i < n) { g_am[i] = 0ull; g_ap[i] = 0ull; g_any[i] = 0u; g_hasin[i] = 0u; g_cnt[i] = 0u; }
}

// ---------------- K1: per-(b,m) global reductions over anchors ----------------
__global__ void __launch_bounds__(256)
k1_metrics(const float* __restrict__ pd_bboxes, const float* __restrict__ anc_points,
           const float* __restrict__ gt_bboxes, const unsigned char* __restrict__ mask_gt,
           unsigned long long* __restrict__ g_am, unsigned long long* __restrict__ g_ap,
           unsigned* __restrict__ g_any, unsigned* __restrict__ g_hasin, int A, int M) {
  int b = blockIdx.y;
  int a = blockIdx.x * blockDim.x + threadIdx.x;
  __shared__ float s_g[MAXM][4];
  __shared__ unsigned s_mgt[MAXM], s_any[MAXM], s_hin[MAXM];
  __shared__ unsigned long long s_am[MAXM], s_ap[MAXM];
  int t = threadIdx.x;
  if (t < M) {
    size_t o = (size_t)b * M + t;
    const float* gp = gt_bboxes + o * 4;
    s_g[t][0] = gp[0]; s_g[t][1] = gp[1]; s_g[t][2] = gp[2]; s_g[t][3] = gp[3];
    s_mgt[t] = mask_gt[o] ? 1u : 0u;
    s_any[t] = 0u; s_hin[t] = 0u; s_am[t] = 0ull; s_ap[t] = 0ull;
  }
  __syncthreads();
  if (a < A) {
    const float* pb = pd_bboxes + ((size_t)b * A + a) * 4;
    float px1 = pb[0], py1 = pb[1], px2 = pb[2], py2 = pb[3];
    float ax = anc_points[(size_t)a * 2], ay = anc_points[(size_t)a * 2 + 1];
    for (int m = 0; m < M; ++m) {
      float gx1 = s_g[m][0], gy1 = s_g[m][1], gx2 = s_g[m][2], gy2 = s_g[m][3];
      float ov = ov_iou(px1, py1, px2, py2, gx1, gy1, gx2, gy2);
      float mn = fminf(fminf(ax - gx1, ay - gy1), fminf(gx2 - ax, gy2 - ay));
      bool ing = mn > 1e-9f;
      if (ing) { s_hin[m] = 1u; if (s_mgt[m]) s_any[m] = 1u; }
      atomicMax(&s_am[m], packmax(ing ? ov : 0.f, a));   // overlaps*mask_in_gts argmax
      atomicMax(&s_ap[m], packmax(ov, a));               // overlaps argmax
    }
  }
  __syncthreads();
  if (t < M) {
    size_t o = (size_t)b * M + t;
    if (s_any[t]) atomicOr(&g_any[o], 1u);
    if (s_hin[t]) atomicOr(&g_hasin[o], 1u);
    atomicMax(&g_am[o], s_am[t]);
    atomicMax(&g_ap[o], s_ap[t]);
  }
}

// ---------------- K2: fix-1 + first select_highest_overlaps; bitmask + counts ----------------
__global__ void __launch_bounds__(256)
k2_select(const float* __restrict__ pd_bboxes, const float* __restrict__ anc_points,
          const float* __restrict__ gt_bboxes, const unsigned char* __restrict__ mask_gt,
          const unsigned long long* __restrict__ g_am, const unsigned long long* __restrict__ g_ap,
          const unsigned* __restrict__ g_any, const unsigned* __restrict__ g_hasin,
          unsigned* __restrict__ g_mp, unsigned* __restrict__ g_cnt, int A, int M) {
  int b = blockIdx.y;
  int a = blockIdx.x * blockDim.x + threadIdx.x;
  __shared__ float s_g[MAXM][4];
  __shared__ unsigned s_mgt[MAXM], s_need[MAXM], s_cnt[MAXM];
  __shared__ int s_best1[MAXM];
  int t = threadIdx.x;
  if (t < M) {
    size_t o = (size_t)b * M + t;
    const float* gp = gt_bboxes + o * 4;
    s_g[t][0] = gp[0]; s_g[t][1] = gp[1]; s_g[t][2] = gp[2]; s_g[t][3] = gp[3];
    unsigned mgt = mask_gt[o] ? 1u : 0u;
    s_mgt[t] = mgt;
    s_need[t] = (mgt && g_any[o] == 0u) ? 1u : 0u;
    s_best1[t] = unpack_anchor(g_hasin[o] ? g_am[o] : g_ap[o]);
    s_cnt[t] = 0u;
  }
  __syncthreads();
  if (a < A) {
    const float* pb = pd_bboxes + ((size_t)b * A + a) * 4;
    float px1 = pb[0], py1 = pb[1], px2 = pb[2], py2 = pb[3];
    float ax = anc_points[(size_t)a * 2], ay = anc_points[(size_t)a * 2 + 1];
    unsigned bits = 0u;
    float best = -1.f; int bidx = 0;
    for (int m = 0; m < M; ++m) {
      float gx1 = s_g[m][0], gy1 = s_g[m][1], gx2 = s_g[m][2], gy2 = s_g[m][3];
      float ov = ov_iou(px1, py1, px2, py2, gx1, gy1, gx2, gy2);
      float mn = fminf(fminf(ax - gx1, ay - gy1), fminf(gx2 - ax, gy2 - ay));
      bool ing = mn > 1e-9f;
      if (ov > best) { best = ov; bidx = m; }   // first-max tie-break (ascending, strict >)
      bool pos = (ing && s_mgt[m]) || (s_need[m] && s_best1[m] == a);
      if (pos) bits |= (1u << m);
    }
    int fg = __popc(bits);
    int tg = 0;
    if (fg > 1) { bits = (1u << bidx); tg = bidx; }
    else if (fg == 1) tg = __ffs(bits) - 1;
    g_mp[(size_t)b * A + a] = bits;
    if (bits) atomicAdd(&s_cnt[tg], 1u);
  }
  __syncthreads();
  if (t < M) atomicAdd(&g_cnt[(size_t)b * M + t], s_cnt[t]);
}

// ---------------- K3: sequential per-gt scan (exact lax.scan semantics) ----------------
__global__ void k3_scan(const unsigned char* __restrict__ mask_gt,
                        const unsigned long long* __restrict__ g_ap,
                        const unsigned* __restrict__ g_cnt,
                        const unsigned* __restrict__ g_mp,
                        int* __restrict__ g_upd, int A, int M, int B) {
  int b = blockIdx.x * blockDim.x + threadIdx.x;
  if (b >= B) return;
  int cnt[MAXM], bp[MAXM];
  for (int g = 0; g < M; ++g) {
    size_t o = (size_t)b * M + g;
    cnt[g] = (int)g_cnt[o];
    bp[g] = unpack_anchor(g_ap[o]);  // argmax_a overlaps[:, :, g]
  }
  int aid[MAXM], atg[MAXM], afg[MAXM]; int nt = 0;  // tracked best-anchor table
  for (int g = 0; g < M; ++g) {
    size_t o = (size_t)b * M + g;
    bool need = (mask_gt[o] != 0) && (cnt[g] == 0);
    int upd = -1;
    if (need) {
      int as = bp[g];
      int idx = -1;
      for (int i = 0; i < nt; ++i) if (aid[i] == as) { idx = i; break; }
      if (idx < 0) {
        unsigned bits = g_mp[(size_t)b * A + as];
        aid[nt] = as; afg[nt] = bits ? 1 : 0; atg[nt] = bits ? (__ffs(bits) - 1) : 0;
        idx = nt++;
      }
      if (afg[idx]) cnt[atg[idx]]--;   // anchor's tgt overwritten -> old gt loses it
      atg[idx] = g; afg[idx] = 1; cnt[g]++;
      upd = as;
    }
    g_upd[o] = upd;
  }
}

// ---------------- K4: WMMA class-score gather + final select + outputs ----------------
__global__ void __launch_bounds__(256)
k4_final(const float* __restrict__ pd_scores, const float* __restrict__ pd_bboxes,
         const float* __restrict__ gt_bboxes, const int* __restrict__ gt_labels,
         const unsigned* __restrict__ g_mp, const int* __restrict__ g_upd,
         float* __restrict__ out, int A, int M, int C, int B) {
  int b = blockIdx.y;
  int tileBase = blockIdx.x * 128;           // 128 anchors per block, 16 per wave
  __shared__ float s_g[MAXM][4];
  __shared__ int s_lab[MAXM], s_upd[MAXM];
  __shared__ float s_cls[8][16 * 33];        // [wave][anchorRow*33 + gt], padded
  int tid = threadIdx.x;
  if (tid < M) {
    size_t o = (size_t)b * M + tid;
    const float* gp = gt_bboxes + o * 4;
    s_g[tid][0] = gp[0]; s_g[tid][1] = gp[1]; s_g[tid][2] = gp[2]; s_g[tid][3] = gp[3];
    s_lab[tid] = gt_labels[o];
    s_upd[tid] = g_upd[o];
  }
  __syncthreads();

  // --- GEMM: cls(16x32) = scores_tile(16xC) x onehot(labels)(Cx32), f32 exact ---
  int lane = tid & 31, wave = tid >> 5;
  int row = lane & 15, kh = lane >> 4;       // A frag: M = lane&15, K pair = 2*(lane>>4)
  int aRow = tileBase + wave * 16 + row;
  if (aRow >= A) aRow = A - 1;               // clamp: keep EXEC all-ones for WMMA
  const float* srow = pd_scores + ((size_t)b * A + aRow) * (size_t)C;
  __builtin_prefetch(srow, 0, 1);            // global_prefetch_b8
  int l0 = s_lab[row];
  int l1 = (M > 16) ? s_lab[row + 16] : -1;
  v8f acc0 = {}; v8f acc1 = {};
  for (int k0 = 0; k0 < C; k0 += 4) {
    int k = k0 + 2 * kh;
    v2f af; af.x = srow[k]; af.y = srow[k + 1];
    v2f b0; b0.x = (l0 == k) ? 1.f : 0.f; b0.y = (l0 == k + 1) ? 1.f : 0.f;
    v2f b1; b1.x = (l1 == k) ? 1.f : 0.f; b1.y = (l1 == k + 1) ? 1.f : 0.f;
    acc0 = __builtin_amdgcn_wmma_f32_16x16x4_f32(false, af, false, b0, (short)0, acc0, false, false);
    acc1 = __builtin_amdgcn_wmma_f32_16x16x4_f32(false, af, false, b1, (short)0, acc1, false, false);
  }
#pragma unroll
  for (int r = 0; r < 8; ++r) {              // D: row = r + 8*(lane>>4), col = lane&15
    int ar = r + 8 * kh;
    s_cls[wave][ar * 33 + row] = acc0[r];
    s_cls[wave][ar * 33 + 16 + row] = acc1[r];
  }
  __syncthreads();

  // --- zero-fill ts for this tile (coalesced) ---
  size_t BA = (size_t)B * A;
  float* ts = out + BA * 5;
  int nA = A - tileBase; if (nA > 128) nA = 128;
  if (nA > 0) {
    size_t tsBase = ((size_t)b * A + tileBase) * (size_t)C;
    for (int i = tid; i < nA * C; i += 256) ts[tsBase + i] = 0.f;
  }

  // --- per-anchor finalize (one thread per anchor, tid < 128) ---
  int myA = tileBase + tid;
  int myTl = 0; float val = 0.f; bool doScatter = false;
  if (tid < 128 && myA < A) {
    size_t oa = (size_t)b * A + myA;
    unsigned bits = g_mp[oa];
    for (int g = 0; g < M; ++g) if (s_upd[g] == myA) bits |= (1u << g);
    const float* pb = pd_bboxes + oa * 4;
    float px1 = pb[0], py1 = pb[1], px2 = pb[2], py2 = pb[3];
    int tz = bits ? (__ffs(bits) - 1) : 0;
    float best = -1.f; int bidx = 0; float ovTz = 0.f;
    for (int m = 0; m < M; ++m) {
      float ov = ov_iou(px1, py1, px2, py2, s_g[m][0], s_g[m][1], s_g[m][2], s_g[m][3]);
      if (ov > best) { best = ov; bidx = m; }
      if (m == tz) ovTz = ov;
    }
    int fg = __popc(bits);
    int tgt; float ovt; int fgo;
    if (fg > 1)      { tgt = bidx; ovt = best; fgo = 1; }   // dedup: one-hot(argmax ov)
    else if (fg == 1){ tgt = tz;   ovt = ovTz; fgo = 1; }
    else             { tgt = 0;    ovt = 0.f;  fgo = 0; }
    float cls = s_cls[tid >> 4][(tid & 15) * 33 + tgt];     // WMMA-gathered pd_scores[b,a,label[tgt]]
    float ov2 = ovt * ovt;
    float align = cls * (ov2 * ov2 * ov2);                  // cls^1 * ov^6
    float norm = fgo ? (align * ovt / (align + 1e-9f)) : 0.f;
    int tl = s_lab[tgt];
    tl = tl < 0 ? 0 : (tl > TAA_NUM_CLASSES ? TAA_NUM_CLASSES : tl);
    out[oa] = (float)tl;                                    // target_labels
    float* tb = out + BA + oa * 4;                          // target_bboxes
    tb[0] = s_g[tgt][0]; tb[1] = s_g[tgt][1]; tb[2] = s_g[tgt][2]; tb[3] = s_g[tgt][3];
    out[BA * 85 + oa] = fgo ? 1.f : 0.f;                    // fg_mask
    out[BA * 86 + oa] = (float)tgt;                         // target_gt_idx
    if (tl < C) { myTl = tl; val = norm; doScatter = true; }
  }
  __syncthreads();
  if (doScatter)
    ts[((size_t)b * A + myA) * (size_t)C + myTl] = val;     // one-hot * norm
}

extern "C" void kernel_launch(void* const* d_in, const int* in_sizes, int n_in,
                              void* d_out, int out_size, void* d_ws, size_t ws_size,
                              hipStream_t stream) {
  const float* pd_scores = (const float*)d_in[0];
  const float* pd_bboxes = (const float*)d_in[1];
  const float* anc_points = (const float*)d_in[2];
  const int* gt_labels = (const int*)d_in[3];
  const float* gt_bboxes = (const float*)d_in[4];
  const unsigned char* mask_gt = (const unsigned char*)d_in[5];

  int A = in_sizes[2] / 2;
  int B = in_sizes[1] / (A * 4);
  int M = in_sizes[3] / B;          // <= 32 (bitmask representation)
  int C = in_sizes[0] / (B * A);    // 80
  int BM = B * M;

  char* ws = (char*)d_ws;
  size_t off = 0;
  unsigned long long* g_am = (unsigned long long*)(ws + off); off += (size_t)BM * 8;
  unsigned long long* g_ap = (unsigned long long*)(ws + off); off += (size_t)BM * 8;
  unsigned* g_any   = (unsigned*)(ws + off); off += (size_t)BM * 4;
  unsigned* g_hasin = (unsigned*)(ws + off); off += (size_t)BM * 4;
  unsigned* g_cnt   = (unsigned*)(ws + off); off += (size_t)BM * 4;
  int*      g_upd   = (int*)(ws + off);      off += (size_t)BM * 4;
  unsigned* g_mp    = (unsigned*)(ws + off); off += (size_t)B * A * 4;

  k0_init<<<(BM + 255) / 256, 256, 0, stream>>>(g_am, g_ap, g_any, g_hasin, g_cnt, BM);
  dim3 g1((A + 255) / 256, B);
  k1_metrics<<<g1, 256, 0, stream>>>(pd_bboxes, anc_points, gt_bboxes, mask_gt,
                                     g_am, g_ap, g_any, g_hasin, A, M);
  k2_select<<<g1, 256, 0, stream>>>(pd_bboxes, anc_points, gt_bboxes, mask_gt,
                                    g_am, g_ap, g_any, g_hasin, g_mp, g_cnt, A, M);
  k3_scan<<<(B + 63) / 64, 64, 0, stream>>>(mask_gt, g_ap, g_cnt, g_mp, g_upd, A, M, B);
  dim3 g4((A + 127) / 128, B);
  k4_final<<<g4, 256, 0, stream>>>(pd_scores, pd_bboxes, gt_bboxes, gt_labels,
                                   g_mp, g_upd, (float*)d_out, A, M, C, B);
}